// TopPLogitProcessor_9483287790146
// MI455X (gfx1250) — compile-verified
//
#include <hip/hip_runtime.h>
#include <float.h>
#include <math.h>
#include <stdint.h>

#define TPB    512
#define NBINS  65536
#define BPT    (NBINS / TPB)   // 128 bins per thread in the hierarchical scan
#define NWAVES (TPB / 32)      // 16 waves (wave32)

// Monotone float -> u32 ordered key (total order, -inf..+inf ascending).
__device__ __forceinline__ unsigned okey(float x) {
  unsigned u = __float_as_uint(x);
  return (u & 0x80000000u) ? ~u : (u | 0x80000000u);
}

// Combine two (running-max, scaled-sum) softmax partials.
__device__ __forceinline__ void mcombine(float& m, float& s, float mo, float so) {
  float M2 = fmaxf(m, mo);
  s = s * __expf(m - M2) + so * __expf(mo - M2);
  m = M2;
}

// CDNA5 async copy: 16B/lane global -> LDS, tracked by ASYNCcnt.
// vdst = LDS byte offset (u32), vaddr = 64-bit global address, GV mode.
__device__ __forceinline__ void async_copy16(const void* g, unsigned lds_off) {
  asm volatile("global_load_async_to_lds_b128 %0, %1, off"
               :: "v"(lds_off), "v"(g) : "memory");
}
__device__ __forceinline__ void wait_async_le1() {
  asm volatile("s_wait_asynccnt 1" ::: "memory");
}
__device__ __forceinline__ void wait_async_le0() {
  asm volatile("s_wait_asynccnt 0" ::: "memory");
}

__global__ __launch_bounds__(TPB)
void topp_mask_kernel(const float* __restrict__ logits,
                      float* __restrict__ out, int V) {
  extern __shared__ unsigned char smem[];

  const int tid = threadIdx.x;
  const float* row  = logits + (size_t)blockIdx.x * (size_t)V;
  float*       orow = out    + (size_t)blockIdx.x * (size_t)V;
  const int V4 = V >> 2;
  const float4* row4 = (const float4*)row;
  float4*       out4 = (float4*)orow;

  // LDS layout (time-multiplexed):
  //   [0, 256KB)         bins[65536] f32   (radix rounds)
  //   [0, 16KB)          stage[2*TPB] f4   (pass-1 async staging; overlaps bins)
  //   [256KB, +2KB)      chunk[TPB] f32    (scan scratch; also wave-reduce slots)
  //   [258KB, +32B)      broadcast slots
  float*  bins  = (float*)smem;
  float4* stage = (float4*)smem;
  float*  chunk = (float*)(smem + (size_t)NBINS * 4);
  float*  bcf   = (float*)(smem + (size_t)NBINS * 4 + TPB * 4);
  unsigned* bcu = (unsigned*)bcf;

  // ---------- Pass 1: online max M and Z = sum exp(x - M), async-staged ----------
  float m = -FLT_MAX, s = 0.f;
  if (tid < V4)
    async_copy16(row4 + tid, (unsigned)(uintptr_t)&stage[tid]);
  int buf = 0;
  for (int i = tid; i < V4; i += TPB) {
    if (i + TPB < V4) {               // wave-uniform for V=128000, TPB=512
      async_copy16(row4 + (i + TPB),
                   (unsigned)(uintptr_t)&stage[(buf ^ 1) * TPB + tid]);
      wait_async_le1();               // oldest (current buf) complete: in-order
    } else {
      wait_async_le0();
    }
    float4 v = stage[buf * TPB + tid];
    float vm = fmaxf(fmaxf(v.x, v.y), fmaxf(v.z, v.w));
    if (vm > m) { s *= __expf(m - vm); m = vm; }
    s += __expf(v.x - m) + __expf(v.y - m) + __expf(v.z - m) + __expf(v.w - m);
    buf ^= 1;
  }
  for (int j = 4 * V4 + tid; j < V; j += TPB) {   // scalar tail (V%4)
    float x = row[j];
    if (x > m) { s *= __expf(m - x); m = x; }
    s += __expf(x - m);
  }

  // Block reduction of (m, s): wave32 shuffle tree, then cross-wave via LDS.
  const int lane = tid & 31, wid = tid >> 5;
  for (int off = 16; off > 0; off >>= 1) {
    float mo = __shfl_xor(m, off, 32);
    float so = __shfl_xor(s, off, 32);
    mcombine(m, s, mo, so);
  }
  float* wm = chunk;                 // 16 slots
  float* ws = chunk + NWAVES;        // 16 slots
  if (lane == 0) { wm[wid] = m; ws[wid] = s; }
  __syncthreads();
  if (wid == 0) {
    float mm = (lane < NWAVES) ? wm[lane] : -FLT_MAX;
    float ss = (lane < NWAVES) ? ws[lane] : 0.f;
    for (int off = 16; off > 0; off >>= 1) {
      float mo = __shfl_xor(mm, off, 32);
      float so = __shfl_xor(ss, off, 32);
      mcombine(mm, ss, mo, so);
    }
    // top_p threshold: float(1.0 - 0.9) rounds to 0.1f (matches JAX promotion)
    if (lane == 0) { bcf[0] = mm; bcf[1] = 0.1f * ss; }
  }
  __syncthreads();
  const float M = bcf[0];
  float Tcur = bcf[1];               // remaining droppable mass

  // ---------- Radix-select: 2 rounds of 16-bit digits, exp-weighted ----------
  // Find smallest key K with mass(keys <= K) > Tcur; keep iff okey(x) >= K.
  unsigned prefix = 0;
  for (int round = 0; round < 2; ++round) {
    for (int i = tid; i < NBINS; i += TPB) bins[i] = 0.f;
    __syncthreads();
    const int shift = (round == 0) ? 16 : 0;
    for (int i = tid; i < V4; i += TPB) {
      float4 v = row4[i];
      float xs[4] = {v.x, v.y, v.z, v.w};
#pragma unroll
      for (int c = 0; c < 4; ++c) {
        unsigned k = okey(xs[c]);
        if (round == 0 || (k >> 16) == prefix)
          atomicAdd(&bins[(k >> shift) & 0xFFFFu], __expf(xs[c] - M)); // ds_add_f32
      }
    }
    for (int j = 4 * V4 + tid; j < V; j += TPB) {
      float x = row[j];
      unsigned k = okey(x);
      if (round == 0 || (k >> 16) == prefix)
        atomicAdd(&bins[(k >> shift) & 0xFFFFu], __expf(x - M));
    }
    __syncthreads();

    // Hierarchical scan: 512 parallel chunk sums, then one thread scans 512+128.
    float cs = 0.f;
    const int base = tid * BPT;
    for (int j = 0; j < BPT; ++j) cs += bins[base + j];
    chunk[tid] = cs;
    __syncthreads();
    if (tid == 0) {
      float c = 0.f;
      int ci = 0;
      for (; ci < TPB - 1; ++ci) {
        float nb = chunk[ci];
        if (c + nb > Tcur) break;
        c += nb;
      }
      int j = 0;
      for (; j < BPT - 1; ++j) {
        float nb = bins[ci * BPT + j];
        if (c + nb > Tcur) break;
        c += nb;
      }
      bcu[2] = (unsigned)(ci * BPT + j);
      bcf[3] = Tcur - c;
    }
    __syncthreads();
    unsigned d = bcu[2];
    Tcur = bcf[3];
    prefix = (round == 0) ? d : ((prefix << 16) | d);
    __syncthreads();
  }
  const unsigned K = prefix;         // exact 32-bit boundary key

  // ---------- Final pass: mask ----------
  const float NEG_INF = -__builtin_inff();
  for (int i = tid; i < V4; i += TPB) {
    float4 v = row4[i];
    float4 o;
    o.x = (okey(v.x) >= K) ? v.x : NEG_INF;
    o.y = (okey(v.y) >= K) ? v.y : NEG_INF;
    o.z = (okey(v.z) >= K) ? v.z : NEG_INF;
    o.w = (okey(v.w) >= K) ? v.w : NEG_INF;
    out4[i] = o;
  }
  for (int j = 4 * V4 + tid; j < V; j += TPB) {
    float x = row[j];
    orow[j] = (okey(x) >= K) ? x : NEG_INF;
  }
}

extern "C" void kernel_launch(void* const* d_in, const int* in_sizes, int n_in,
                              void* d_out, int out_size, void* d_ws, size_t ws_size,
                              hipStream_t stream) {
  (void)d_ws; (void)ws_size; (void)out_size;
  const float* logits = (const float*)d_in[0];
  int B = (n_in > 1 && in_sizes[1] > 0) ? in_sizes[1] : 128;   // position_ids len = batch
  int V = in_sizes[0] / B;                                     // 128000
  float* out = (float*)d_out;
  // Dynamic LDS: 256KB bins + 2KB scan scratch + 32B broadcast (<= 320KB/WGP on CDNA5)
  size_t smem = (size_t)NBINS * 4 + (size_t)TPB * 4 + 32;
  hipLaunchKernelGGL(topp_mask_kernel, dim3(B), dim3(TPB), smem, stream,
                     logits, out, V);
}